// ResidualSeparableBlock_71820443124025
// MI455X (gfx1250) — compile-verified
//
#include <hip/hip_runtime.h>
#include <hip/hip_bf16.h>

// ---------------------------------------------------------------------------
// ResidualSeparableBlock for CDNA5 (gfx1250, wave32)
//
//   out = LN2(relu( Σ_k T2_k @ Q2_k + b2 )) + maps
//   h1  = LN1(relu( Σ_k T1_k @ Q1_k + b1 ))
//   T_k = Chebyshev recursion of sparse graph Laplacian (8 / 20 neighbors)
//   Q_k[f,o] = W[k,f,0]*P[2f,o] + W[k,f,1]*P[2f+1,o]   (folded depthwise+pointwise)
//
// GEMMs use V_WMMA_F32_16X16X4_F32 (exact fp32), 4 independent accumulator
// chains per wave. One fused kernel launch per Chebyshev step (global
// dependency through the gather). All T buffers are L2-resident (25 MB each
// vs 192 MB L2).
// ---------------------------------------------------------------------------

#define N_PIX 49152
#define NBATCH 2
#define FDIM 64
#define TOTAL_ROWS (N_PIX * NBATCH)          // 98304 (b*N + n)
#define WAVES_PER_BLOCK 8
#define ROWS_PER_WAVE 16
#define ROWS_PER_BLOCK (WAVES_PER_BLOCK * ROWS_PER_WAVE)   // 128
#define STEP_BLOCKS (TOTAL_ROWS / ROWS_PER_BLOCK)          // 768
#define LDS_STRIDE 66                        // 16x64 tile padded: kills bank conflicts

typedef float v2f __attribute__((ext_vector_type(2)));
typedef float v8f __attribute__((ext_vector_type(8)));

// --------------------------- Q_k precompute --------------------------------
__global__ __launch_bounds__(256)
void build_q_kernel(const float* __restrict__ W, const float* __restrict__ P,
                    float* __restrict__ Q, int K)
{
    int t = blockIdx.x * 256 + threadIdx.x;
    if (t >= K * FDIM * FDIM) return;
    int o = t & 63;
    int f = (t >> 6) & 63;
    int k = t >> 12;
    Q[t] = W[(k * 64 + f) * 2 + 0] * P[(2 * f) * 64 + o]
         + W[(k * 64 + f) * 2 + 1] * P[(2 * f + 1) * 64 + o];
}

// ------------------ fused Chebyshev step + WMMA accumulate -----------------
// APPLY=1 :  Tnext = c1*(d*Tcur + Σ_j w_j*Tcur[idx_j]) - c0*Tprev  (written
//            to global AND staged in LDS)
// APPLY=0 :  tile = Tcur (k=0 GEMM-only path; Tnext untouched)
// then    :  acc += tile @ Q   (INIT=1: acc = tile @ Q + bias)
template<int NB, bool APPLY, bool INIT>
__global__ __launch_bounds__(256)
void cheb_step_kernel(const float* __restrict__ Tcur,
                      const float* __restrict__ Tprev,
                      float* __restrict__ Tnext,
                      const int*   __restrict__ idx,
                      const float* __restrict__ wgt,
                      const float* __restrict__ dvec,
                      const float* __restrict__ Q,     // [64,64] for this k
                      const float* __restrict__ bias,  // [64]
                      float* __restrict__ acc,         // [TOTAL_ROWS,64]
                      float c1, float c0)
{
    __shared__ float lds[WAVES_PER_BLOCK][ROWS_PER_WAVE * LDS_STRIDE];

    const int wave = threadIdx.x >> 5;
    const int lane = threadIdx.x & 31;
    const int g0   = (blockIdx.x * WAVES_PER_BLOCK + wave) * ROWS_PER_WAVE;
    float* tile = lds[wave];
    const int f0 = lane * 2;                 // each lane owns 2 features

    if (APPLY) {
        const int b      = g0 / N_PIX;       // 16 consecutive rows share b
        const int basebn = b * N_PIX;
        for (int r = 0; r < ROWS_PER_WAVE; ++r) {
            const int g = g0 + r;
            const int n = g - basebn;
            const float dn = dvec[n];
            const float2 xv = *(const float2*)(Tcur + g * FDIM + f0);
            float a0 = dn * xv.x, a1 = dn * xv.y;
            const int*   ip = idx + (size_t)n * NB;
            const float* wp = wgt + (size_t)n * NB;
            #pragma unroll
            for (int j = 0; j < NB; ++j) {
                const int   nj = ip[j];
                const float wj = wp[j];
                const float2 nv = *(const float2*)(Tcur + (basebn + nj) * FDIM + f0);
                a0 += wj * nv.x;
                a1 += wj * nv.y;
            }
            float o0 = c1 * a0, o1 = c1 * a1;
            if (c0 != 0.0f) {
                const float2 pv = *(const float2*)(Tprev + g * FDIM + f0);
                o0 -= c0 * pv.x;
                o1 -= c0 * pv.y;
            }
            tile[r * LDS_STRIDE + f0]     = o0;
            tile[r * LDS_STRIDE + f0 + 1] = o1;
            *(float2*)(Tnext + g * FDIM + f0) = make_float2(o0, o1);
        }
    } else {
        #pragma unroll
        for (int r = 0; r < ROWS_PER_WAVE; ++r) {
            const float2 xv = *(const float2*)(Tcur + (g0 + r) * FDIM + f0);
            tile[r * LDS_STRIDE + f0]     = xv.x;
            tile[r * LDS_STRIDE + f0 + 1] = xv.y;
        }
    }
    __syncthreads();

    // ---- 16x64 @ 64x64 via V_WMMA_F32_16X16X4_F32, 4 independent chains ----
    // A 16x4 frag: lane<16 -> {K0,K1} of row M=lane ; lane>=16 -> {K2,K3}
    const int half = lane >> 4;              // 0: K0/K1 & rows 0-7 ; 1: K2/K3 & rows 8-15
    const int lr   = lane & 15;
    v2f afr[16];
    #pragma unroll
    for (int c = 0; c < 16; ++c) {
        afr[c].x = tile[lr * LDS_STRIDE + c * 4 + 2 * half];
        afr[c].y = tile[lr * LDS_STRIDE + c * 4 + 2 * half + 1];
    }
    v8f cf[4];
    #pragma unroll
    for (int t = 0; t < 4; ++t) {            // four 16-wide output column tiles
        const int col = t * 16 + lr;
        if (INIT) {
            const float bv = bias[col];
            #pragma unroll
            for (int r = 0; r < 8; ++r) cf[t][r] = bv;
        } else {
            #pragma unroll
            for (int r = 0; r < 8; ++r)
                cf[t][r] = acc[(g0 + r + 8 * half) * FDIM + col];
        }
    }
    #pragma unroll
    for (int c = 0; c < 16; ++c) {
        #pragma unroll
        for (int t = 0; t < 4; ++t) {
            // B 4x16 frag: lane<16 -> rows {K0,K1} col=lr ; lane>=16 -> {K2,K3}
            const int col = t * 16 + lr;
            v2f bf;
            bf.x = Q[(c * 4 + 2 * half) * FDIM + col];
            bf.y = Q[(c * 4 + 2 * half + 1) * FDIM + col];
            cf[t] = __builtin_amdgcn_wmma_f32_16x16x4_f32(
                        false, afr[c], false, bf, (short)0, cf[t], false, false);
        }
    }
    #pragma unroll
    for (int t = 0; t < 4; ++t) {
        const int col = t * 16 + lr;
        #pragma unroll
        for (int r = 0; r < 8; ++r)
            acc[(g0 + r + 8 * half) * FDIM + col] = cf[t][r];
    }
}

// --------------------------- ReLU + LayerNorm ------------------------------
__global__ __launch_bounds__(256)
void relu_ln_kernel(const float* __restrict__ in,     // acc [rows,64]
                    const float* __restrict__ s,
                    const float* __restrict__ bb,
                    const float* __restrict__ resid,   // nullptr or [rows,64]
                    float* __restrict__ out)
{
    const int row  = blockIdx.x * 8 + (threadIdx.x >> 5);
    const int lane = threadIdx.x & 31;
    const float* p = in + (size_t)row * FDIM;
    float x0 = fmaxf(p[lane], 0.0f);
    float x1 = fmaxf(p[lane + 32], 0.0f);
    float sum = x0 + x1;
    float sq  = x0 * x0 + x1 * x1;
    for (int m = 16; m >= 1; m >>= 1) {
        sum += __shfl_xor(sum, m, 32);
        sq  += __shfl_xor(sq,  m, 32);
    }
    const float mu  = sum * (1.0f / FDIM);
    const float var = sq * (1.0f / FDIM) - mu * mu;
    const float inv = rsqrtf(var + 1e-5f);
    float y0 = (x0 - mu) * inv * s[lane]      + bb[lane];
    float y1 = (x1 - mu) * inv * s[lane + 32] + bb[lane + 32];
    if (resid) {
        y0 += resid[(size_t)row * FDIM + lane];
        y1 += resid[(size_t)row * FDIM + lane + 32];
    }
    out[(size_t)row * FDIM + lane]      = y0;
    out[(size_t)row * FDIM + lane + 32] = y1;
}

// ------------------------------- host side ---------------------------------
template<int NB, bool APPLY, bool INIT>
static inline void launch_step(const float* cur, const float* prev, float* next,
                               const int* idx, const float* wgt, const float* dvec,
                               const float* Qk, const float* bias, float* acc,
                               float c1, float c0, hipStream_t stream)
{
    cheb_step_kernel<NB, APPLY, INIT><<<dim3(STEP_BLOCKS), dim3(256), 0, stream>>>(
        cur, prev, next, idx, wgt, dvec, Qk, bias, acc, c1, c0);
}

extern "C" void kernel_launch(void* const* d_in, const int* in_sizes, int n_in,
                              void* d_out, int out_size, void* d_ws, size_t ws_size,
                              hipStream_t stream)
{
    const float* maps  = (const float*)d_in[0];
    const int*   idx8  = (const int*)  d_in[1];
    const float* w8    = (const float*)d_in[2];
    const float* d8    = (const float*)d_in[3];
    const int*   idx20 = (const int*)  d_in[4];
    const float* w20   = (const float*)d_in[5];
    const float* d20   = (const float*)d_in[6];
    const float* W1    = (const float*)d_in[7];
    const float* P1    = (const float*)d_in[8];
    const float* b1    = (const float*)d_in[9];
    const float* W2    = (const float*)d_in[10];
    const float* P2    = (const float*)d_in[11];
    const float* b2    = (const float*)d_in[12];
    const float* ln1s  = (const float*)d_in[13];
    const float* ln1b  = (const float*)d_in[14];
    const float* ln2s  = (const float*)d_in[15];
    const float* ln2b  = (const float*)d_in[16];

    float* acc  = (float*)d_out;                       // [rows,64] accumulator
    const size_t S = (size_t)TOTAL_ROWS * FDIM;        // 6291456 floats
    float* bufA = (float*)d_ws;
    float* bufB = bufA + S;
    float* bufC = bufB + S;
    float* Q1   = bufC + S;                            // 6*64*64
    float* Q2   = Q1 + 6 * FDIM * FDIM;                // 10*64*64

    build_q_kernel<<<(6 * 4096 + 255) / 256, 256, 0, stream>>>(W1, P1, Q1, 6);
    build_q_kernel<<<(10 * 4096 + 255) / 256, 256, 0, stream>>>(W2, P2, Q2, 10);

    // ---------------- layer 1: K=6, 8 neighbors ----------------
    launch_step<8,false,true >(maps, maps, bufA, idx8, w8, d8, Q1 + 0*4096, b1, acc, 0.f, 0.f, stream); // T0
    launch_step<8,true ,false>(maps, maps, bufA, idx8, w8, d8, Q1 + 1*4096, b1, acc, 1.f, 0.f, stream); // T1
    launch_step<8,true ,false>(bufA, maps, bufB, idx8, w8, d8, Q1 + 2*4096, b1, acc, 2.f, 1.f, stream); // T2
    launch_step<8,true ,false>(bufB, bufA, bufC, idx8, w8, d8, Q1 + 3*4096, b1, acc, 2.f, 1.f, stream); // T3
    launch_step<8,true ,false>(bufC, bufB, bufA, idx8, w8, d8, Q1 + 4*4096, b1, acc, 2.f, 1.f, stream); // T4
    launch_step<8,true ,false>(bufA, bufC, bufB, idx8, w8, d8, Q1 + 5*4096, b1, acc, 2.f, 1.f, stream); // T5
    relu_ln_kernel<<<TOTAL_ROWS / 8, 256, 0, stream>>>(acc, ln1s, ln1b, nullptr, bufC); // h1 -> bufC

    // ---------------- layer 2: K=10, 20 neighbors --------------
    launch_step<20,false,true >(bufC, bufC, bufA, idx20, w20, d20, Q2 + 0*4096, b2, acc, 0.f, 0.f, stream); // T0=h1
    launch_step<20,true ,false>(bufC, bufC, bufA, idx20, w20, d20, Q2 + 1*4096, b2, acc, 1.f, 0.f, stream); // T1
    launch_step<20,true ,false>(bufA, bufC, bufB, idx20, w20, d20, Q2 + 2*4096, b2, acc, 2.f, 1.f, stream); // T2
    launch_step<20,true ,false>(bufB, bufA, bufC, idx20, w20, d20, Q2 + 3*4096, b2, acc, 2.f, 1.f, stream); // T3
    launch_step<20,true ,false>(bufC, bufB, bufA, idx20, w20, d20, Q2 + 4*4096, b2, acc, 2.f, 1.f, stream); // T4
    launch_step<20,true ,false>(bufA, bufC, bufB, idx20, w20, d20, Q2 + 5*4096, b2, acc, 2.f, 1.f, stream); // T5
    launch_step<20,true ,false>(bufB, bufA, bufC, idx20, w20, d20, Q2 + 6*4096, b2, acc, 2.f, 1.f, stream); // T6
    launch_step<20,true ,false>(bufC, bufB, bufA, idx20, w20, d20, Q2 + 7*4096, b2, acc, 2.f, 1.f, stream); // T7
    launch_step<20,true ,false>(bufA, bufC, bufB, idx20, w20, d20, Q2 + 8*4096, b2, acc, 2.f, 1.f, stream); // T8
    launch_step<20,true ,false>(bufB, bufA, bufC, idx20, w20, d20, Q2 + 9*4096, b2, acc, 2.f, 1.f, stream); // T9
    relu_ln_kernel<<<TOTAL_ROWS / 8, 256, 0, stream>>>(acc, ln2s, ln2b, maps, (float*)d_out);
}